// EdgeModel_4329327035190
// MI455X (gfx1250) — compile-verified
//
#include <hip/hip_runtime.h>

#define N_EDGES 500000
#define N_NODES 50000
#define D_NODE  128
#define D_EDGE  64
#define D_IN    320
#define D_HID   256
#define D_OUT   64

// K-chunk of W1 staged in LDS per iteration
#define KC 32
#define NKC (D_IN / KC)          // 10
#define W1T_STRIDE 34            // transposed W1 chunk row stride (pad, even for b64 align)
#define HS_STRIDE  260           // h tile row stride (pad, even for b64 align)

typedef float v2f __attribute__((ext_vector_type(2)));
typedef float v4f __attribute__((ext_vector_type(4)));
typedef float v8f __attribute__((ext_vector_type(8)));

static __device__ __forceinline__ v8f zero8() {
    v8f z = {0.f, 0.f, 0.f, 0.f, 0.f, 0.f, 0.f, 0.f};
    return z;
}

__global__ __launch_bounds__(256)
void edge_mlp_fused(const float* __restrict__ node_feats,
                    const long long* __restrict__ edge_index,
                    const float* __restrict__ edge_attr,
                    const float* __restrict__ W1,
                    const float* __restrict__ b1,
                    const float* __restrict__ W2,
                    const float* __restrict__ b2,
                    float* __restrict__ out)
{
    extern __shared__ float smem[];
    float* W1t = smem;    // [D_HID][W1T_STRIDE] transposed chunk (34,816 B)
    float* Hs  = smem;    // union after phase 1: [64][HS_STRIDE]   (66,560 B)

    const int tid   = threadIdx.x;
    const int lane  = tid & 31;
    const int wid   = tid >> 5;
    const int m     = wid & 3;        // M sub-tile (16 rows each)
    const int nh    = wid >> 2;       // N half of hidden dim
    const int l15   = lane & 15;
    const int khalf = (lane >> 4) << 1;  // 0 for lanes 0-15, 2 for lanes 16-31

    // hint: W2 will be needed later; warm L2
    __builtin_prefetch(W2 + tid, 0, 0);

    // per-lane gathered row base pointers (clamped for tail tile; EXEC stays full)
    const int ebase = blockIdx.x * 64;
    const int lr    = ebase + m * 16 + l15;
    const int e     = lr < N_EDGES ? lr : N_EDGES - 1;
    const int nrow  = (int)edge_index[e];
    const int ncol  = (int)edge_index[N_EDGES + e];
    const float* pA = node_feats + (long long)nrow * D_NODE;
    const float* pB = node_feats + (long long)ncol * D_NODE;
    const float* pC = edge_attr  + (long long)e    * D_EDGE;

    // ---------------- Phase 1: h = relu(x @ W1 + b1), strip 16 x 128 ----------------
    v8f acc[8];
    #pragma unroll
    for (int n = 0; n < 8; ++n) acc[n] = zero8();

    for (int kc = 0; kc < NKC; ++kc) {
        __syncthreads();
        // stage W1[kc*KC .. kc*KC+31][0..255] transposed into LDS, coalesced b128 reads
        #pragma unroll
        for (int i = 0; i < 8; ++i) {
            const int idx = i * 1024 + tid * 4;       // 0..8191, step 4
            const int k   = idx >> 8;                 // 0..31
            const int c   = idx & 255;                // multiple of 4
            v4f v = *(const v4f*)(W1 + (long long)(kc * KC + k) * D_HID + c);
            W1t[(c + 0) * W1T_STRIDE + k] = v.x;
            W1t[(c + 1) * W1T_STRIDE + k] = v.y;
            W1t[(c + 2) * W1T_STRIDE + k] = v.z;
            W1t[(c + 3) * W1T_STRIDE + k] = v.w;
        }
        __syncthreads();

        for (int ks = 0; ks < 8; ++ks) {
            const int k0   = kc * KC + ks * 4;
            const int koff = k0 + khalf;
            // select gather source region for this k-step (region uniform per step)
            const float* src = (koff < 128) ? (pA + koff)
                             : (koff < 256) ? (pB + (koff - 128))
                                            : (pC + (koff - 256));
            const v2f a = *(const v2f*)src;          // global_load_b64 (L2 hit)
            const int kl = ks * 4 + khalf;
            #pragma unroll
            for (int n = 0; n < 8; ++n) {
                const int cc = nh * 128 + n * 16 + l15;
                const v2f b = *(const v2f*)(W1t + cc * W1T_STRIDE + kl);  // ds_load_b64
                acc[n] = __builtin_amdgcn_wmma_f32_16x16x4_f32(
                    false, a, false, b, (short)0, acc[n], false, false);
            }
        }
    }

    __syncthreads();   // all waves done reading W1t; reuse smem as Hs

    // bias + ReLU, write h tile to LDS (C layout: reg j -> M=j (lanes<16) / M=8+j (lanes>=16))
    {
        const int rb = m * 16 + ((lane >> 4) << 3);
        #pragma unroll
        for (int n = 0; n < 8; ++n) {
            const int cc = nh * 128 + n * 16 + l15;
            const float bias = b1[cc];
            #pragma unroll
            for (int j = 0; j < 8; ++j) {
                float h = acc[n][j] + bias;
                h = h > 0.f ? h : 0.f;
                Hs[(rb + j) * HS_STRIDE + cc] = h;
            }
        }
    }
    __syncthreads();

    // ---------------- Phase 2: out = h @ W2 + b2, strip 16 x 32 ----------------
    v8f o0 = zero8(), o1 = zero8();
    const int arow = m * 16 + l15;
    for (int kk = 0; kk < D_HID / 4; ++kk) {
        const int koff = kk * 4 + khalf;
        const v2f a = *(const v2f*)(Hs + arow * HS_STRIDE + koff);  // ds_load_b64
        {
            const int cc = nh * 32 + l15;
            v2f b;
            b.x = W2[(long long)koff * D_OUT + cc];
            b.y = W2[(long long)(koff + 1) * D_OUT + cc];
            o0 = __builtin_amdgcn_wmma_f32_16x16x4_f32(
                false, a, false, b, (short)0, o0, false, false);
        }
        {
            const int cc = nh * 32 + 16 + l15;
            v2f b;
            b.x = W2[(long long)koff * D_OUT + cc];
            b.y = W2[(long long)(koff + 1) * D_OUT + cc];
            o1 = __builtin_amdgcn_wmma_f32_16x16x4_f32(
                false, a, false, b, (short)0, o1, false, false);
        }
    }

    // epilogue: bias + guarded store
    {
        const int rb = m * 16 + ((lane >> 4) << 3);
        #pragma unroll
        for (int n = 0; n < 2; ++n) {
            const int cc = nh * 32 + n * 16 + l15;
            const float bias = b2[cc];
            const v8f ov = (n == 0) ? o0 : o1;
            #pragma unroll
            for (int j = 0; j < 8; ++j) {
                const int er = ebase + rb + j;
                if (er < N_EDGES)
                    out[(long long)er * D_OUT + cc] = ov[j] + bias;
            }
        }
    }
}

extern "C" void kernel_launch(void* const* d_in, const int* in_sizes, int n_in,
                              void* d_out, int out_size, void* d_ws, size_t ws_size,
                              hipStream_t stream) {
    (void)in_sizes; (void)n_in; (void)d_ws; (void)ws_size; (void)out_size;
    const float*     node_feats = (const float*)d_in[0];
    const long long* edge_index = (const long long*)d_in[1];  // int64 in reference
    const float*     edge_attr  = (const float*)d_in[2];
    const float*     W1         = (const float*)d_in[3];
    const float*     b1         = (const float*)d_in[4];
    const float*     W2         = (const float*)d_in[5];
    const float*     b2         = (const float*)d_in[6];
    float*           out        = (float*)d_out;

    const int grid = (N_EDGES + 63) / 64;                 // 7813 workgroups
    const size_t shmem = (size_t)64 * HS_STRIDE * sizeof(float);  // 66,560 B (covers W1t too)
    edge_mlp_fused<<<grid, 256, shmem, stream>>>(node_feats, edge_index, edge_attr,
                                                 W1, b1, W2, b2, out);
}